// ImprovedMarginRankingLoss_21157008900111
// MI455X (gfx1250) — compile-verified
//
#include <hip/hip_runtime.h>

// ---------------------------------------------------------------------------
// ImprovedMarginRankingLoss for MI455X (gfx1250, wave32)
//   out = mean( relu(0.2 - |x - mean(x)|) ),  N = 33554432 f32
// Two streaming passes (pass 2 hits the 192MB L2: dataset is 134MB).
// f32 reductions run on the matrix pipe: V_WMMA_F32_16X16X4_F32 with a
// ones(16x4) A-matrix => one WMMA = 64-element f32 sum into the accumulator.
// Two independent accumulators break the D->C dependency chain.
// ---------------------------------------------------------------------------

#define N_TOTAL   33554432u
#define MARGIN_F  0.2f
#define BLOCKS    1024
#define THREADS   256
#define WAVES_PER_BLOCK (THREADS / 32)

typedef float v2f __attribute__((ext_vector_type(2)));
typedef float v8f __attribute__((ext_vector_type(8)));

#if __has_builtin(__builtin_amdgcn_wmma_f32_16x16x4_f32)
#define HAVE_WMMA_F32 1
#endif

// Accumulate 64 f32 values (2 per lane in b, across the wave) into c.
// A = ones(16x4)  =>  D[m][n] = sum_k B[k][n] + C[m][n]; every row identical.
// Extraction rule: wave_sum = 0.5 * sum_over_32_lanes(c[0])
//   (VGPR0 of C/D holds row M=0 in lanes 0-15 and row M=8 in lanes 16-31,
//    both equal to the 16 column sums -> lane-sum double counts by exactly 2x)
__device__ __forceinline__ v8f acc64(v8f c, v2f b) {
#ifdef HAVE_WMMA_F32
    const v2f ones = {1.0f, 1.0f};
    // 8 args: (neg_a, A, neg_b, B, c_mod, C, reuse_a, reuse_b)
    return __builtin_amdgcn_wmma_f32_16x16x4_f32(
        false, ones, false, b, (short)0, c, false, false);
#else
    // VALU fallback preserving the same extraction rule (factor 2)
    c[0] += 2.0f * (b[0] + b[1]);
    return c;
#endif
}

// Deterministic wave32 butterfly + per-block LDS combine. Valid in thread 0.
__device__ __forceinline__ float block_reduce(float v, float* smem) {
#pragma unroll
    for (int off = 16; off > 0; off >>= 1)
        v += __shfl_xor(v, off, 32);
    const unsigned lane = threadIdx.x & 31u;
    const unsigned wave = threadIdx.x >> 5;
    if (lane == 0) smem[wave] = v;
    __syncthreads();
    float r = 0.0f;
    if (threadIdx.x == 0) {
#pragma unroll
        for (int w = 0; w < WAVES_PER_BLOCK; ++w) r += smem[w];
    }
    return r;
}

// ------------------------------ Pass 1: sum(x) ------------------------------
__global__ void __launch_bounds__(THREADS)
sum_pass(const float* __restrict__ in, float* __restrict__ partials) {
    const float4* __restrict__ in4 = (const float4*)in;
    const unsigned n4     = N_TOTAL / 4u;
    const unsigned tid    = blockIdx.x * THREADS + threadIdx.x;
    const unsigned stride = gridDim.x * THREADS;

    v8f c0 = {0.f, 0.f, 0.f, 0.f, 0.f, 0.f, 0.f, 0.f};
    v8f c1 = c0;
    float tail = 0.0f;

    // Main body: uniform trip count across the wave -> EXEC all ones for WMMA.
    const unsigned iters  = n4 / stride;        // 32 for this problem size
    const unsigned iters4 = (iters / 4u) * 4u;
    unsigned i = tid;
    for (unsigned k = 0; k < iters4; k += 4u) { // 4-deep load batching for MLP
        float4 v0 = in4[i];
        float4 v1 = in4[i + stride];
        float4 v2 = in4[i + 2u * stride];
        float4 v3 = in4[i + 3u * stride];
        i += 4u * stride;
        c0 = acc64(c0, (v2f){v0.x, v0.y});  c1 = acc64(c1, (v2f){v0.z, v0.w});
        c0 = acc64(c0, (v2f){v1.x, v1.y});  c1 = acc64(c1, (v2f){v1.z, v1.w});
        c0 = acc64(c0, (v2f){v2.x, v2.y});  c1 = acc64(c1, (v2f){v2.z, v2.w});
        c0 = acc64(c0, (v2f){v3.x, v3.y});  c1 = acc64(c1, (v2f){v3.z, v3.w});
    }
    for (unsigned k = iters4; k < iters; ++k) {
        float4 v = in4[i]; i += stride;
        c0 = acc64(c0, (v2f){v.x, v.y});    c1 = acc64(c1, (v2f){v.z, v.w});
    }
    // Residual (possibly divergent) tail: plain VALU adds, no WMMA.
    for (unsigned j = tid + iters * stride; j < n4; j += stride) {
        float4 v = in4[j];
        tail += (v.x + v.y) + (v.z + v.w);
    }

    __shared__ float smem[WAVES_PER_BLOCK];
    float r = block_reduce(0.5f * (c0[0] + c1[0]) + tail, smem);
    if (threadIdx.x == 0) partials[blockIdx.x] = r;
}

// ----------------------- Pass 2: sum(relu(m - |x-mu|)) ----------------------
__global__ void __launch_bounds__(THREADS)
loss_pass(const float* __restrict__ in, const float* __restrict__ mean_ptr,
          float* __restrict__ partials) {
    const float mu = mean_ptr[0];               // uniform -> scalar load
    const float4* __restrict__ in4 = (const float4*)in;
    const unsigned n4     = N_TOTAL / 4u;
    const unsigned tid    = blockIdx.x * THREADS + threadIdx.x;
    const unsigned stride = gridDim.x * THREADS;

    v8f c0 = {0.f, 0.f, 0.f, 0.f, 0.f, 0.f, 0.f, 0.f};
    v8f c1 = c0;
    float tail = 0.0f;

    const unsigned iters  = n4 / stride;
    const unsigned iters4 = (iters / 4u) * 4u;
    unsigned i = tid;
#define LOSS(s) fmaxf(MARGIN_F - fabsf((s) - mu), 0.0f)
    for (unsigned k = 0; k < iters4; k += 4u) {
        float4 v0 = in4[i];
        float4 v1 = in4[i + stride];
        float4 v2 = in4[i + 2u * stride];
        float4 v3 = in4[i + 3u * stride];
        i += 4u * stride;
        c0 = acc64(c0, (v2f){LOSS(v0.x), LOSS(v0.y)});
        c1 = acc64(c1, (v2f){LOSS(v0.z), LOSS(v0.w)});
        c0 = acc64(c0, (v2f){LOSS(v1.x), LOSS(v1.y)});
        c1 = acc64(c1, (v2f){LOSS(v1.z), LOSS(v1.w)});
        c0 = acc64(c0, (v2f){LOSS(v2.x), LOSS(v2.y)});
        c1 = acc64(c1, (v2f){LOSS(v2.z), LOSS(v2.w)});
        c0 = acc64(c0, (v2f){LOSS(v3.x), LOSS(v3.y)});
        c1 = acc64(c1, (v2f){LOSS(v3.z), LOSS(v3.w)});
    }
    for (unsigned k = iters4; k < iters; ++k) {
        float4 v = in4[i]; i += stride;
        c0 = acc64(c0, (v2f){LOSS(v.x), LOSS(v.y)});
        c1 = acc64(c1, (v2f){LOSS(v.z), LOSS(v.w)});
    }
    for (unsigned j = tid + iters * stride; j < n4; j += stride) {
        float4 v = in4[j];
        tail += (LOSS(v.x) + LOSS(v.y)) + (LOSS(v.z) + LOSS(v.w));
    }
#undef LOSS

    __shared__ float smem[WAVES_PER_BLOCK];
    float r = block_reduce(0.5f * (c0[0] + c1[0]) + tail, smem);
    if (threadIdx.x == 0) partials[blockIdx.x] = r;
}

// ------------------- Final deterministic single-block reduces ---------------
__global__ void __launch_bounds__(BLOCKS)
reduce_final(const float* __restrict__ partials, float* __restrict__ out,
             float scale) {
    __shared__ float smem[BLOCKS];
    smem[threadIdx.x] = partials[threadIdx.x];
    __syncthreads();
#pragma unroll
    for (int s = BLOCKS / 2; s > 0; s >>= 1) {
        if ((int)threadIdx.x < s) smem[threadIdx.x] += smem[threadIdx.x + s];
        __syncthreads();
    }
    if (threadIdx.x == 0) out[0] = smem[0] * scale;
}

// ---------------------------------------------------------------------------
extern "C" void kernel_launch(void* const* d_in, const int* in_sizes, int n_in,
                              void* d_out, int out_size, void* d_ws, size_t ws_size,
                              hipStream_t stream) {
    (void)in_sizes; (void)n_in; (void)out_size; (void)ws_size;
    const float* in  = (const float*)d_in[0];
    float*       out = (float*)d_out;

    float* ws        = (float*)d_ws;
    float* partials1 = ws;                  // BLOCKS floats
    float* mean_buf  = ws + BLOCKS;         // 1 float
    float* partials2 = ws + BLOCKS + 1;     // BLOCKS floats

    sum_pass   <<<BLOCKS, THREADS, 0, stream>>>(in, partials1);
    reduce_final<<<1, BLOCKS, 0, stream>>>(partials1, mean_buf,
                                           1.0f / (float)N_TOTAL);
    loss_pass  <<<BLOCKS, THREADS, 0, stream>>>(in, mean_buf, partials2);
    reduce_final<<<1, BLOCKS, 0, stream>>>(partials2, out,
                                           1.0f / (float)N_TOTAL);
}